// Aggregator_87239375716568
// MI455X (gfx1250) — compile-verified
//
#include <hip/hip_runtime.h>

#define N_USERS    50000
#define N_ITEMS    50000
#define N_ENTITIES 100000
#define N_REL      32
#define E_EDGES    2000000
#define NI_PAIRS   1000000
#define DIM        64

typedef __attribute__((ext_vector_type(2))) float v2f;
typedef __attribute__((ext_vector_type(8))) float v8f;

// ---------------------------------------------------------------------------
// zero fill
// ---------------------------------------------------------------------------
__global__ __launch_bounds__(256) void zero_kernel(float* __restrict__ p, int n) {
    int i = blockIdx.x * 256 + threadIdx.x;
    if (i < n) p[i] = 0.0f;
}

// ---------------------------------------------------------------------------
// KG edge scatter: one wave32 per edge, 2 dims per lane.
// sum[head] += entity_emb[tail] * weight[edge_type]; cnt[head] += 1
// ---------------------------------------------------------------------------
__global__ __launch_bounds__(256) void kg_scatter_kernel(
    const float* __restrict__ ent, const int* __restrict__ ei,
    const int* __restrict__ et, const float* __restrict__ w,
    float* __restrict__ esum, float* __restrict__ ecnt) {
    int tid  = blockIdx.x * 256 + threadIdx.x;
    int e    = tid >> 5;
    if (e >= E_EDGES) return;
    int lane = tid & 31;
    int head = ei[e];               // per-wave uniform
    int tail = ei[E_EDGES + e];
    int r    = et[e];
    float2 v  = *(const float2*)(ent + (size_t)tail * DIM + lane * 2);
    float2 wv = *(const float2*)(w + r * DIM + lane * 2);
    float* dst = esum + (size_t)head * DIM + lane * 2;
    atomicAdd(dst + 0, v.x * wv.x);
    atomicAdd(dst + 1, v.y * wv.y);
    if (lane == 0) atomicAdd(ecnt + head, 1.0f);
}

// ---------------------------------------------------------------------------
// entity mean; item rows stay in ws (input to the WMMA gate), attribute rows
// (i >= N_ITEMS) go straight to the entity section of d_out.
// ---------------------------------------------------------------------------
__global__ __launch_bounds__(256) void mean_split_kernel(
    float* __restrict__ esum, const float* __restrict__ ecnt,
    float* __restrict__ ent_out) {
    int tid = blockIdx.x * 256 + threadIdx.x;
    int i   = tid >> 5;
    if (i >= N_ENTITIES) return;
    int lane = tid & 31;
    float c = fmaxf(ecnt[i], 1.0f);
    float inv = 1.0f / c;
    float2* p = (float2*)(esum + (size_t)i * DIM + lane * 2);
    float2 v = *p;
    v.x *= inv; v.y *= inv;
    if (i < N_ITEMS) {
        *p = v;                                  // item_kg_agg (ws)
    } else {
        *(float2*)(ent_out + (size_t)i * DIM + lane * 2) = v;  // att_kg_agg
    }
}

// ---------------------------------------------------------------------------
// user->item scatter: sum[mat_col] += user_emb[mat_row] * weight[0]
// ---------------------------------------------------------------------------
__global__ __launch_bounds__(256) void iu_scatter_kernel(
    const float* __restrict__ uemb, const int* __restrict__ mrow,
    const int* __restrict__ mcol, const float* __restrict__ w0,
    float* __restrict__ isum, float* __restrict__ icnt) {
    int tid = blockIdx.x * 256 + threadIdx.x;
    int e   = tid >> 5;
    if (e >= NI_PAIRS) return;
    int lane = tid & 31;
    int r = mrow[e];
    int c = mcol[e];
    float2 v  = *(const float2*)(uemb + (size_t)r * DIM + lane * 2);
    float2 wv = *(const float2*)(w0 + lane * 2);
    float* dst = isum + (size_t)c * DIM + lane * 2;
    atomicAdd(dst + 0, v.x * wv.x);
    atomicAdd(dst + 1, v.y * wv.y);
    if (lane == 0) atomicAdd(icnt + c, 1.0f);
}

__global__ __launch_bounds__(256) void iu_mean_kernel(
    float* __restrict__ isum, const float* __restrict__ icnt) {
    int tid = blockIdx.x * 256 + threadIdx.x;
    int i   = tid >> 5;
    if (i >= N_ITEMS) return;
    int lane = tid & 31;
    float inv = 1.0f / fmaxf(icnt[i], 1.0f);
    float2* p = (float2*)(isum + (size_t)i * DIM + lane * 2);
    float2 v = *p;
    v.x *= inv; v.y *= inv;
    *p = v;
}

// ---------------------------------------------------------------------------
// Gate GEMM + fusion, one wave32 per 16x16 output tile using
// V_WMMA_F32_16X16X4_F32, accumulating over K=64 for both gate matrices:
//   Z = item_kg_agg @ G1^T + i_u_agg @ G2^T
//   gi = sigmoid(Z);  fusion = gi*item_kg + (1-gi)*i_u  -> d_out item rows
// f32 A 16x4 layout: lane L(<16): a = A[L][k0..k0+1]; lane L(>=16): A[L-16][k0+2..3]
// f32 B 4x16 layout mirrored; B[k][j] = gateW[j*64 + k] (reference applies .T)
// C/D layout: VGPR v: lanes 0-15 -> (M=v, N=lane), lanes 16-31 -> (M=v+8, N=lane-16)
// ---------------------------------------------------------------------------
__global__ __launch_bounds__(32) void gate_fusion_kernel(
    const float* __restrict__ item_kg,  // N_ITEMS x 64 (ws, post-mean)
    const float* __restrict__ iu,       // N_ITEMS x 64 (ws, post-mean)
    const float* __restrict__ g1,       // 64 x 64 row-major (j, k)
    const float* __restrict__ g2,       // 64 x 64 row-major (j, k)
    float* __restrict__ out_fusion) {   // d_out entity section, rows < N_ITEMS
    const int tile = blockIdx.x;
    const int tM = tile >> 2;           // 0 .. N_ITEMS/16-1
    const int tN = tile & 3;            // 0 .. 3 (64/16)
    const int lane = threadIdx.x;
    const int hi = lane >> 4;           // 0: lanes 0-15, 1: lanes 16-31
    const int l15 = lane & 15;

    const int rowA = tM * 16 + l15;     // A-matrix row (M = lane&15)
    const int colB = tN * 16 + l15;     // B-matrix col (output feature j)

    v8f acc = {};
    #pragma unroll
    for (int k0 = 0; k0 < DIM; k0 += 4) {
        const int ka = k0 + hi * 2;
        v2f a1, b1, a2, b2;
        a1.x = item_kg[rowA * DIM + ka];
        a1.y = item_kg[rowA * DIM + ka + 1];
        b1.x = g1[colB * DIM + ka];
        b1.y = g1[colB * DIM + ka + 1];
        acc = __builtin_amdgcn_wmma_f32_16x16x4_f32(
            false, a1, false, b1, (short)0, acc, false, false);
        a2.x = iu[rowA * DIM + ka];
        a2.y = iu[rowA * DIM + ka + 1];
        b2.x = g2[colB * DIM + ka];
        b2.y = g2[colB * DIM + ka + 1];
        acc = __builtin_amdgcn_wmma_f32_16x16x4_f32(
            false, a2, false, b2, (short)0, acc, false, false);
    }

    const int ncol = tN * 16 + l15;     // output feature for this lane
    #pragma unroll
    for (int v = 0; v < 8; ++v) {
        const int m = tM * 16 + v + hi * 8;   // item row for acc[v]
        float z  = acc[v];
        float gi = 1.0f / (1.0f + __expf(-z));
        float a  = item_kg[m * DIM + ncol];
        float b  = iu[m * DIM + ncol];
        out_fusion[m * DIM + ncol] = gi * a + (1.0f - gi) * b;
    }
}

// ---------------------------------------------------------------------------
// user_agg[mat_row] += item_emb_fusion[mat_col]   (segment_sum)
// ---------------------------------------------------------------------------
__global__ __launch_bounds__(256) void user_scatter_kernel(
    const float* __restrict__ fusion, const int* __restrict__ mrow,
    const int* __restrict__ mcol, float* __restrict__ uout) {
    int tid = blockIdx.x * 256 + threadIdx.x;
    int e   = tid >> 5;
    if (e >= NI_PAIRS) return;
    int lane = tid & 31;
    int r = mrow[e];
    int c = mcol[e];
    float2 v = *(const float2*)(fusion + (size_t)c * DIM + lane * 2);
    float* dst = uout + (size_t)r * DIM + lane * 2;
    atomicAdd(dst + 0, v.x);
    atomicAdd(dst + 1, v.y);
}

// ---------------------------------------------------------------------------
extern "C" void kernel_launch(void* const* d_in, const int* in_sizes, int n_in,
                              void* d_out_v, int out_size, void* d_ws, size_t ws_size,
                              hipStream_t stream) {
    const float* entity_emb = (const float*)d_in[0];
    const float* user_emb   = (const float*)d_in[1];
    const int*   edge_index = (const int*)d_in[2];
    const int*   edge_type  = (const int*)d_in[3];
    const int*   mat_row    = (const int*)d_in[4];
    const int*   mat_col    = (const int*)d_in[5];
    // d_in[6] = mat_val (all ones, unused by the reference math)
    const float* weight     = (const float*)d_in[7];
    const float* gate1_w    = (const float*)d_in[8];
    const float* gate2_w    = (const float*)d_in[9];
    float* out = (float*)d_out_v;

    // workspace layout (floats): 9,750,000 total (~39 MB)
    float* ws      = (float*)d_ws;
    float* ent_sum = ws;                                    // N_ENTITIES*64
    float* ent_cnt = ent_sum + (size_t)N_ENTITIES * DIM;    // N_ENTITIES
    float* iu_sum  = ent_cnt + N_ENTITIES;                  // N_ITEMS*64
    float* iu_cnt  = iu_sum + (size_t)N_ITEMS * DIM;        // N_ITEMS

    float* ent_out  = out;                                  // entity_agg_out
    float* user_out = out + (size_t)N_ENTITIES * DIM;       // user_agg

    // 0) zero accumulators + user_agg output (every call: deterministic)
    const int n_ws = N_ENTITIES * DIM + N_ENTITIES + N_ITEMS * DIM + N_ITEMS;
    zero_kernel<<<(n_ws + 255) / 256, 256, 0, stream>>>(ws, n_ws);
    const int n_uo = N_USERS * DIM;
    zero_kernel<<<(n_uo + 255) / 256, 256, 0, stream>>>(user_out, n_uo);

    // 1) KG edge scatter-sum (+counts)
    {
        long long t = (long long)E_EDGES * 32;
        kg_scatter_kernel<<<(unsigned)((t + 255) / 256), 256, 0, stream>>>(
            entity_emb, edge_index, edge_type, weight, ent_sum, ent_cnt);
    }
    // 2) entity mean; att rows -> d_out, item rows stay in ws
    {
        long long t = (long long)N_ENTITIES * 32;
        mean_split_kernel<<<(unsigned)((t + 255) / 256), 256, 0, stream>>>(
            ent_sum, ent_cnt, ent_out);
    }
    // 3) user->item scatter-sum (+counts), weight row 0
    {
        long long t = (long long)NI_PAIRS * 32;
        iu_scatter_kernel<<<(unsigned)((t + 255) / 256), 256, 0, stream>>>(
            user_emb, mat_row, mat_col, weight, iu_sum, iu_cnt);
    }
    // 4) i_u mean (in place)
    {
        long long t = (long long)N_ITEMS * 32;
        iu_mean_kernel<<<(unsigned)((t + 255) / 256), 256, 0, stream>>>(iu_sum, iu_cnt);
    }
    // 5) WMMA gate GEMM + sigmoid fusion -> item rows of d_out
    {
        const int tiles = (N_ITEMS / 16) * (DIM / 16);  // 12500 wave32 tiles
        gate_fusion_kernel<<<tiles, 32, 0, stream>>>(
            ent_sum, iu_sum, gate1_w, gate2_w, ent_out);
    }
    // 6) user_agg segment-sum from fused item embeddings
    {
        long long t = (long long)NI_PAIRS * 32;
        user_scatter_kernel<<<(unsigned)((t + 255) / 256), 256, 0, stream>>>(
            ent_out, mat_row, mat_col, user_out);
    }
}